// QuantumPercolationEngine_47098611368145
// MI455X (gfx1250) — compile-verified
//
#include <hip/hip_runtime.h>
#include <math.h>

#define HID 128
#define INDIM 64
#define MAXDEG 6
#define WPB 8            // waves per block (blockDim = 256)
#define NBLK 1024        // blocks for the cell kernel

// ---------- wave32 reductions (fixed butterfly order -> deterministic) ----------
__device__ __forceinline__ float waveSum(float v) {
#pragma unroll
    for (int o = 16; o > 0; o >>= 1) v += __shfl_xor(v, o, 32);
    return v;
}
__device__ __forceinline__ float waveMax(float v) {
#pragma unroll
    for (int o = 16; o > 0; o >>= 1) v = fmaxf(v, __shfl_xor(v, o, 32));
    return v;
}
__device__ __forceinline__ float4 ld4(const float* p) {
    return *reinterpret_cast<const float4*>(p);
}

// ---------- stage 1: input encode -> pp[128] ----------
__global__ void k_encode(const float* __restrict__ x, const float* __restrict__ W_in,
                         const float* __restrict__ b_in, float* __restrict__ pp) {
    __shared__ float sm[HID];
    const int h = threadIdx.x;            // 128 threads
    float s = b_in[h];
#pragma unroll 8
    for (int k = 0; k < INDIM; ++k) s = fmaf(x[k], W_in[h * INDIM + k], s);
    sm[h] = fabsf(s);
    __syncthreads();
    for (int st = HID / 2; st > 0; st >>= 1) {
        if (h < st) sm[h] = fmaxf(sm[h], sm[h + st]);
        __syncthreads();
    }
    const float m = sm[0];
    pp[h] = s / (m + 1e-8f) * 0.31415926535897931f;   // pi * 0.1
}

// ---------- stage 2: per-cell physics (one wave per cell, grid-stride) ----------
__global__ __launch_bounds__(256) void k_cells(
    const float* __restrict__ amp_, const float* __restrict__ ph_, const float* __restrict__ pv_,
    const int* __restrict__ nidx, const float* __restrict__ nsgn,
    const unsigned char* __restrict__ nmask,
    const float* __restrict__ pp, const int* __restrict__ stepp,
    float* __restrict__ partials, int n)
{
    const int lane = threadIdx.x & 31;
    const int wav  = threadIdx.x >> 5;
    const int gwave = blockIdx.x * WPB + wav;
    const int nwaves = gridDim.x * WPB;
    const float t = 0.1f * (float)stepp[0];
    const float twopi_n = 6.28318530717958647f / (float)n;

    // per-thread partial accumulators (4 channels per lane)
    float accRe[4] = {0.f, 0.f, 0.f, 0.f}, accIm[4] = {0.f, 0.f, 0.f, 0.f};
    float s1 = 0.f, s2 = 0.f;

    float ppa[4];
    {
        float4 p = ld4(pp + lane * 4);
        ppa[0] = p.x; ppa[1] = p.y; ppa[2] = p.z; ppa[3] = p.w;
    }

    for (int i = gwave; i < n; i += nwaves) {
        const int base = i * HID + lane * 4;
        float4 a4 = ld4(amp_ + base);
        float4 p4 = ld4(ph_  + base);
        float4 v4 = ld4(pv_  + base);
        float ai[4] = {a4.x, a4.y, a4.z, a4.w};
        float ci[4], si[4], rei[4], imi[4];
        {
            float th[4] = {p4.x + 0.1f * v4.x, p4.y + 0.1f * v4.y,
                           p4.z + 0.1f * v4.z, p4.w + 0.1f * v4.w};
#pragma unroll
            for (int c = 0; c < 4; ++c) {
                sincosf(th[c], &si[c], &ci[c]);
                rei[c] = ai[c] * ci[c];
                imi[c] = ai[c] * si[c];
            }
        }

        // neighbor metadata (wave-uniform)
        int   idx[MAXDEG]; float sgn[MAXDEG]; int mk[MAXDEG];
        int deg = 0;
#pragma unroll
        for (int d = 0; d < MAXDEG; ++d) {
            idx[d] = nidx[i * MAXDEG + d];
            sgn[d] = nsgn[i * MAXDEG + d];
            mk[d]  = (int)nmask[i * MAXDEG + d];
            deg += mk[d];
        }

        float cRe[4] = {0.f, 0.f, 0.f, 0.f}, cIm[4] = {0.f, 0.f, 0.f, 0.f};
        float bestv = -3.0e38f; int bestj = -1;
#pragma unroll
        for (int d = 0; d < MAXDEG; ++d) {
            if (!mk[d]) continue;                    // wave-uniform branch
            const int j = idx[d];
            const int jb = j * HID + lane * 4;
            float4 aj4 = ld4(amp_ + jb);
            float4 pj4 = ld4(ph_  + jb);
            float4 vj4 = ld4(pv_  + jb);
            float aj[4] = {aj4.x, aj4.y, aj4.z, aj4.w};
            // neighbor strength = sum over 128 channels of amp
            const float strj = waveSum(aj[0] + aj[1] + aj[2] + aj[3]);
            float thj[4] = {pj4.x + 0.1f * vj4.x, pj4.y + 0.1f * vj4.y,
                            pj4.z + 0.1f * vj4.z, pj4.w + 0.1f * vj4.w};
#pragma unroll
            for (int c = 0; c < 4; ++c) {
                float cj, sj;
                sincosf(thj[c], &sj, &cj);
                const float cd = ci[c] * cj + si[c] * sj;   // cos(thi - thj)
                const float w  = cd * sgn[d] * aj[c];
                cRe[c] += w * cj;
                cIm[c] += w * sj;
            }
            if (strj > bestv) { bestv = strj; bestj = j; }   // first-max
        }

        float nre[4], nim[4];
        if (deg > 0) {
            const float k = 0.3f * 0.1f / (float)deg;
#pragma unroll
            for (int c = 0; c < 4; ++c) {
                nre[c] = 0.7f * rei[c] + k * cRe[c];
                nim[c] = 0.7f * imi[c] + k * cIm[c];
            }
        } else {
#pragma unroll
            for (int c = 0; c < 4; ++c) { nre[c] = rei[c]; nim[c] = imi[c]; }
        }

        // morphism toward strongest neighbor: new += 0.02 * (new/|new|) * conj(src)
        if (deg >= 2) {
            const int jb = bestj * HID + lane * 4;
            float4 ab4 = ld4(amp_ + jb);
            float4 pb4 = ld4(ph_  + jb);
            float4 vb4 = ld4(pv_  + jb);
            float ab[4] = {ab4.x, ab4.y, ab4.z, ab4.w};
            float thb[4] = {pb4.x + 0.1f * vb4.x, pb4.y + 0.1f * vb4.y,
                            pb4.z + 0.1f * vb4.z, pb4.w + 0.1f * vb4.w};
#pragma unroll
            for (int c = 0; c < 4; ++c) {
                float cb, sb;
                sincosf(thb[c], &sb, &cb);
                const float sre = ab[c] * cb;
                const float sim = ab[c] * sb;
                const float hyp = sqrtf(nre[c] * nre[c] + nim[c] * nim[c]);
                float ur, ui;
                if (hyp > 0.f) { const float inv = 1.0f / hyp; ur = nre[c] * inv; ui = nim[c] * inv; }
                else           { ur = 1.0f; ui = 0.0f; }      // angle(0) == 0
                nre[c] += 0.02f * (ur * sre + ui * sim);
                nim[c] += 0.02f * (ui * sre - ur * sim);
            }
        }

        // traveling wave
        const float wf = 1.0f + 0.02f * sinf(t + twopi_n * (float)i);
#pragma unroll
        for (int c = 0; c < 4; ++c) { nre[c] *= wf; nim[c] *= wf; }

        // input-driven phase rotation on first 32 cells
        if (i < 32) {
            const float fac = 0.1f / (1.0f + 0.1f * (float)i);
#pragma unroll
            for (int c = 0; c < 4; ++c) {
                float sd, cd;
                sincosf(ppa[c] * fac, &sd, &cd);
                const float r = nre[c] * cd - nim[c] * sd;
                const float m = nre[c] * sd + nim[c] * cd;
                nre[c] = r; nim[c] = m;
            }
        }

        // per-cell amplitude renormalization
        float am[4]; float mx = 0.f;
#pragma unroll
        for (int c = 0; c < 4; ++c) {
            am[c] = sqrtf(nre[c] * nre[c] + nim[c] * nim[c]);
            mx = fmaxf(mx, am[c]);
        }
        mx = waveMax(mx);
        const float sc = 1.0f / (mx + 1e-8f);
#pragma unroll
        for (int c = 0; c < 4; ++c) {
            const float cr = nre[c] * sc, cm = nim[c] * sc, aa = am[c] * sc;
            accRe[c] += cr; accIm[c] += cm;
            s1 += aa; s2 += aa * aa;
        }
    }

    // ---- deterministic block reduction of partials ----
    __shared__ float smre[WPB * HID];
    __shared__ float smim[WPB * HID];
    __shared__ float sms1[WPB], sms2[WPB];
#pragma unroll
    for (int c = 0; c < 4; ++c) {
        smre[wav * HID + lane * 4 + c] = accRe[c];
        smim[wav * HID + lane * 4 + c] = accIm[c];
    }
    const float w1 = waveSum(s1), w2 = waveSum(s2);
    if (lane == 0) { sms1[wav] = w1; sms2[wav] = w2; }
    __syncthreads();

    float* P = partials + (size_t)blockIdx.x * 258;
    const int tt = threadIdx.x;
    if (tt < HID) {
        float s = 0.f;
#pragma unroll
        for (int w = 0; w < WPB; ++w) s += smre[w * HID + tt];
        P[tt] = s;
    } else {
        const int hh = tt - HID;
        float s = 0.f;
#pragma unroll
        for (int w = 0; w < WPB; ++w) s += smim[w * HID + hh];
        P[HID + hh] = s;
    }
    if (tt == 0) {
        float a = 0.f, b = 0.f;
#pragma unroll
        for (int w = 0; w < WPB; ++w) { a += sms1[w]; b += sms2[w]; }
        P[256] = a; P[257] = b;
    }
}

// ---------- stage 3: deterministic final reduction ----------
__global__ void k_reduce(const float* __restrict__ partials, int nblocks, int n,
                         float* __restrict__ out_real, float* __restrict__ d_out) {
    __shared__ double sd1[256], sd2[256];
    const int t = threadIdx.x;    // 256 threads
    double a1 = 0.0, a2 = 0.0;
    for (int b = t; b < nblocks; b += 256) {
        a1 += (double)partials[(size_t)b * 258 + 256];
        a2 += (double)partials[(size_t)b * 258 + 257];
    }
    sd1[t] = a1; sd2[t] = a2;

    if (t < HID) {
        double s = 0.0;
        for (int b = 0; b < nblocks; ++b) s += (double)partials[(size_t)b * 258 + t];
        out_real[t] = (float)(s / (double)n);
    } else {
        const int h = t - HID;
        double s = 0.0;
        for (int b = 0; b < nblocks; ++b) s += (double)partials[(size_t)b * 258 + HID + h];
        out_real[HID + h] = (float)(s / (double)n);
    }
    __syncthreads();
    if (t == 0) {
        double S1 = 0.0, S2 = 0.0;
        for (int k = 0; k < 256; ++k) { S1 += sd1[k]; S2 += sd2[k]; }
        const double M = (double)n * (double)HID;
        const double var = (S2 - S1 * S1 / M) / (M - 1.0);   // ddof=1
        d_out[64] = (float)var;
    }
}

// ---------- stage 4: decode GEMV via fp32 WMMA (V_WMMA_F32_16X16X4_F32) ----------
typedef __attribute__((ext_vector_type(2))) float v2f;
typedef __attribute__((ext_vector_type(8))) float v8f;

__global__ void k_decode(const float* __restrict__ or_, const float* __restrict__ W_dec,
                         const float* __restrict__ b_dec, float* __restrict__ pred) {
    // exactly one wave32: EXEC all-1s as required by WMMA
    const int lane = threadIdx.x;
    const int lo = lane & 15;        // row within tile (A), N column (B/D)
    const int hi = lane >> 4;        // K-half select: K = v + 2*hi
#pragma unroll
    for (int mt = 0; mt < 4; ++mt) {
        v8f acc = {};
        const int row = mt * 16 + lo;
#pragma unroll 4
        for (int kk = 0; kk < 64; ++kk) {
            const int k0 = kk * 4 + 2 * hi;
            v2f a, b;
            a.x = W_dec[row * 256 + k0];
            a.y = W_dec[row * 256 + k0 + 1];
            b.x = or_[k0];               // vector replicated across all N columns
            b.y = or_[k0 + 1];
            acc = __builtin_amdgcn_wmma_f32_16x16x4_f32(
                false, a, false, b, (short)0, acc, false, false);
        }
        if (lo == 0) {                    // read column N = 0
#pragma unroll
            for (int v = 0; v < 8; ++v) {
                const int r = mt * 16 + v + 8 * hi;   // D: M = v + 8*hi
                pred[r] = acc[v] + b_dec[r];
            }
        }
    }
}

extern "C" void kernel_launch(void* const* d_in, const int* in_sizes, int n_in,
                              void* d_out, int out_size, void* d_ws, size_t ws_size,
                              hipStream_t stream) {
    const float* x     = (const float*)d_in[0];
    const float* amp   = (const float*)d_in[1];
    const float* ph    = (const float*)d_in[2];
    const float* pv    = (const float*)d_in[3];
    const int*   nidx  = (const int*)d_in[4];
    const float* nsgn  = (const float*)d_in[5];
    const unsigned char* nmask = (const unsigned char*)d_in[6];
    const float* W_in  = (const float*)d_in[7];
    const float* b_in  = (const float*)d_in[8];
    const float* W_dec = (const float*)d_in[9];
    const float* b_dec = (const float*)d_in[10];
    const int*   step  = (const int*)d_in[11];

    const int n = in_sizes[1] / HID;      // 65536
    float* out = (float*)d_out;           // [0..63]=pred, [64]=tension
    float* ws  = (float*)d_ws;

    float* pp       = ws;                 // 128 floats
    float* out_real = ws + 128;           // 256 floats
    float* partials = ws + 384;           // NBLK * 258 floats (~1.03 MB)

    k_encode<<<1, 128, 0, stream>>>(x, W_in, b_in, pp);
    k_cells <<<NBLK, 256, 0, stream>>>(amp, ph, pv, nidx, nsgn, nmask, pp, step, partials, n);
    k_reduce<<<1, 256, 0, stream>>>(partials, NBLK, n, out_real, out);
    k_decode<<<1, 32, 0, stream>>>(out_real, W_dec, b_dec, out);
}